// NeRFRenderer_5669356834523
// MI455X (gfx1250) — compile-verified
//
#include <hip/hip_runtime.h>
#include <math.h>

// CDNA5 f32 WMMA operand types: A/B = 16x4 / 4x16 f32 (2 VGPRs), C/D = 16x16 f32 (8 VGPRs)
typedef float v2f __attribute__((ext_vector_type(2)));
typedef float v8f __attribute__((ext_vector_type(8)));

#define NRS   128   // num_steps (coarse)
#define NUP   128   // upsample_steps (fine)
#define NTOT  256

namespace {

__device__ __forceinline__ v8f wmma4(v2f a, v2f b, v8f c) {
  // (neg_a, A, neg_b, B, c_mod, C, reuse_a, reuse_b) -> v_wmma_f32_16x16x4_f32
  return __builtin_amdgcn_wmma_f32_16x16x4_f32(false, a, false, b, (short)0, c,
                                               false, false);
}

__device__ __forceinline__ float softplus_f(float x) {
  return (x > 0.f) ? (x + log1pf(expf(-x))) : log1pf(expf(x));
}
__device__ __forceinline__ float sigmoid_f(float x) { return 1.f / (1.f + expf(-x)); }

// Wave-inclusive prefix product of per-lane chunk value; returns inclusive scan.
__device__ __forceinline__ float wave_incl_scan_mul(float v, int l) {
#pragma unroll
  for (int off = 1; off < 32; off <<= 1) {
    float t = __shfl_up(v, off, 32);
    if (l >= off) v *= t;
  }
  return v;
}
__device__ __forceinline__ float wave_incl_scan_add(float v, int l) {
#pragma unroll
  for (int off = 1; off < 32; off <<= 1) {
    float t = __shfl_up(v, off, 32);
    if (l >= off) v += t;
  }
  return v;
}

// Density MLP over 128 points = 8 WMMA M-tiles of 16.
//   h = relu(X[16,3] * W1[3,64] + b1);  o[16,16] = h * W2[64,16] + b2
// sigma -> softplus(o[:,0]); geo -> o[:,1:16]
__device__ void density_mlp_128(const float* xs, const float* ys, const float* zs,
                                const float* sW1 /*[4][64]*/, const float* sb1,
                                const float* sW2 /*[64][16]*/, const float* sb2,
                                float* hbuf /*[16][65]*/,
                                float* sigma_dst, float* geo_dst /*[.][16]*/,
                                int out_base) {
  const int l  = threadIdx.x;
  const int m  = l & 15;   // A-row (point) / D-column (channel) index per lane
  const int hf = l >> 4;   // half-wave -> K pair {0,1} vs {2,3}
  for (int t = 0; t < 8; ++t) {
    const int p = t * 16 + m;
    v2f a;
    a.x = hf ? zs[p] : xs[p];   // K = 2 : 0
    a.y = hf ? 0.f   : ys[p];   // K = 3(pad) : 1
    v8f h[4];
#pragma unroll
    for (int nt = 0; nt < 4; ++nt) {
      const float bias = sb1[nt * 16 + m];
      v8f c;
#pragma unroll
      for (int r = 0; r < 8; ++r) c[r] = bias;
      v2f b;
      b.x = sW1[(hf * 2 + 0) * 64 + nt * 16 + m];
      b.y = sW1[(hf * 2 + 1) * 64 + nt * 16 + m];
      h[nt] = wmma4(a, b, c);
    }
    // ReLU + stage to LDS as [point][channel] for layer-2 A layout
#pragma unroll
    for (int nt = 0; nt < 4; ++nt)
#pragma unroll
      for (int r = 0; r < 8; ++r)
        hbuf[(r + hf * 8) * 65 + nt * 16 + m] = fmaxf(h[nt][r], 0.f);
    __syncthreads();

    v8f o;
    {
      const float bias2 = sb2[m];
#pragma unroll
      for (int r = 0; r < 8; ++r) o[r] = bias2;
    }
#pragma unroll
    for (int ks = 0; ks < 16; ++ks) {
      v2f a2, b2;
      a2.x = hbuf[m * 65 + ks * 4 + hf * 2 + 0];
      a2.y = hbuf[m * 65 + ks * 4 + hf * 2 + 1];
      b2.x = sW2[(ks * 4 + hf * 2 + 0) * 16 + m];
      b2.y = sW2[(ks * 4 + hf * 2 + 1) * 16 + m];
      o = wmma4(a2, b2, o);
    }
    // D layout: lane holds channel N=m for points M = r + hf*8
#pragma unroll
    for (int r = 0; r < 8; ++r) {
      const int pt = out_base + t * 16 + r + hf * 8;
      const float v = o[r];
      if (m == 0) sigma_dst[pt] = softplus_f(v);
      else        geo_dst[pt * 16 + (m - 1)] = v;
    }
    __syncthreads();
  }
}

__global__ __launch_bounds__(32) void nerf_render_kernel(
    const float* __restrict__ rays_o, const float* __restrict__ rays_d,
    const float* __restrict__ W1, const float* __restrict__ b1,
    const float* __restrict__ W2, const float* __restrict__ b2,
    const float* __restrict__ Wc1, const float* __restrict__ bc1,
    const float* __restrict__ Wc2, const float* __restrict__ bc2,
    float* __restrict__ out) {
  __shared__ float sW1[4 * 64];      // K-padded (row 3 = 0)
  __shared__ float sW2[64 * 16];
  __shared__ float sWc1[20 * 64];    // K-padded (rows 18,19 = 0)
  __shared__ float sWc2[64 * 4];     // compact: only 3 real output cols (+pad)
  __shared__ float sb1[64], sb2v[16], sbc1[64], sbc2v[16];
  __shared__ float xs[NRS], ys[NRS], zs[NRS];
  __shared__ float hbuf[16 * 65];
  __shared__ float sigma_src[NTOT];        // concatenated [coarse|fine] order
  __shared__ float geo[NTOT * 16];         // stride 16, 15 used
  __shared__ float zc[NRS], znew[NUP], zall[NTOT];
  __shared__ int   srci[NTOT];             // merge permutation -> source index
  __shared__ float wts[NTOT], wc[NRS], cdfv[NRS], zmid[NRS];

  const int l  = threadIdx.x;
  const int m  = l & 15;
  const int hf = l >> 4;
  const int ray = blockIdx.x;

  // Kick off cacheline prefetches for the data this block is about to read.
  __builtin_prefetch(&rays_o[ray * 3], 0, 3);
  __builtin_prefetch(&rays_d[ray * 3], 0, 3);
  __builtin_prefetch(&W1[l * 6], 0, 3);
  __builtin_prefetch(&W2[l * 32], 0, 3);
  __builtin_prefetch(&Wc1[l * 36], 0, 3);
  __builtin_prefetch(&Wc2[l * 6], 0, 3);

  // ---- stage MLP weights into LDS (reused by every tile) ----
  for (int i = l; i < 4 * 64; i += 32)  { int r = i >> 6, c = i & 63; sW1[i]  = (r < 3)  ? W1[r * 64 + c]  : 0.f; }
  for (int i = l; i < 64 * 16; i += 32) sW2[i] = W2[i];
  for (int i = l; i < 20 * 64; i += 32) { int r = i >> 6, c = i & 63; sWc1[i] = (r < 18) ? Wc1[r * 64 + c] : 0.f; }
  for (int i = l; i < 64 * 4; i += 32)  { int r = i >> 2, c = i & 3;  sWc2[i] = (c < 3)  ? Wc2[r * 3 + c]  : 0.f; }
  for (int i = l; i < 64; i += 32) { sb1[i] = b1[i]; sbc1[i] = bc1[i]; }
  if (l < 16) { sb2v[l] = b2[l]; sbc2v[l] = (l < 3) ? bc2[l] : 0.f; }
  __syncthreads();

  // ---- ray setup (uniform per wave; redundant across lanes) ----
  const float ox = rays_o[ray * 3 + 0], oy = rays_o[ray * 3 + 1], oz = rays_o[ray * 3 + 2];
  const float dx = rays_d[ray * 3 + 0], dy = rays_d[ray * 3 + 1], dz = rays_d[ray * 3 + 2];
  float near, far;
  {
    float ix = 1.f / dx, iy = 1.f / dy, iz = 1.f / dz;
    float t1x = (-1.f - ox) * ix, t2x = (1.f - ox) * ix;
    float t1y = (-1.f - oy) * iy, t2y = (1.f - oy) * iy;
    float t1z = (-1.f - oz) * iz, t2z = (1.f - oz) * iz;
    near = fmaxf(fmaxf(fminf(t1x, t2x), fminf(t1y, t2y)), fminf(t1z, t2z));
    far  = fminf(fminf(fmaxf(t1x, t2x), fmaxf(t1y, t2y)), fmaxf(t1z, t2z));
    near = fmaxf(near, 0.2f);
    far  = fmaxf(far, near + 1e-6f);
  }
  const float sd = (far - near) / (float)NRS;

  // ---- coarse samples ----
  for (int it = 0; it < NRS / 32; ++it) {
    int i = it * 32 + l;
    float z = near + (far - near) * ((float)i / (float)(NRS - 1));
    zc[i] = z;
    xs[i] = fminf(fmaxf(ox + dx * z, -1.f), 1.f);
    ys[i] = fminf(fmaxf(oy + dy * z, -1.f), 1.f);
    zs[i] = fminf(fmaxf(oz + dz * z, -1.f), 1.f);
  }
  __syncthreads();
  density_mlp_128(xs, ys, zs, sW1, sb1, sW2, sb2v, hbuf, sigma_src, geo, 0);

  // ---- coarse transmittance weights: wave-parallel chunked prefix product ----
  {
    float fa[4], lp[4];                 // alpha, chunk-exclusive prefixes
    float P = 1.f;
#pragma unroll
    for (int j = 0; j < 4; ++j) {
      int i = l * 4 + j;
      float delta = (i < NRS - 1) ? (zc[i + 1] - zc[i]) : sd;
      float alpha = 1.f - expf(-delta * sigma_src[i]);
      fa[j] = alpha;
      lp[j] = P;
      P *= (1.f - alpha + 1e-15f);
    }
    float incl = wave_incl_scan_mul(P, l);
    float excl = __shfl_up(incl, 1, 32);
    if (l == 0) excl = 1.f;
#pragma unroll
    for (int j = 0; j < 4; ++j) wc[l * 4 + j] = fa[j] * excl * lp[j];
  }
  for (int i = l; i < NRS - 1; i += 32) zmid[i] = zc[i] + 0.5f * (zc[i + 1] - zc[i]);
  __syncthreads();

  // ---- CDF: wave-parallel chunked prefix sum (weights = wc[1..126]) ----
  {
    float sv[4];
    float S = 0.f;
#pragma unroll
    for (int j = 0; j < 4; ++j) {
      int idx = l * 4 + j;
      float v = (idx < NRS - 2) ? (wc[idx + 1] + 1e-5f) : 0.f;
      S += v;
      sv[j] = S;                        // chunk-inclusive
    }
    float incl = wave_incl_scan_add(S, l);
    float excl = __shfl_up(incl, 1, 32);
    if (l == 0) excl = 0.f;
    float total = __shfl(incl, 31, 32);
    float inv = 1.f / total;
    if (l == 0) cdfv[0] = 0.f;
#pragma unroll
    for (int j = 0; j < 4; ++j) {
      int idx = l * 4 + j;
      if (idx < NRS - 2) cdfv[idx + 1] = (excl + sv[j]) * inv;
    }
  }
  __syncthreads();

  // ---- inverse-CDF sampling: bins=zmid[127], cdf[127] ----
  {
    const int C = NRS - 1;
    const float u0 = 0.5f / (float)NUP;
    const float du = (1.f - 1.f / (float)NUP) / (float)(NUP - 1);
    for (int it = 0; it < NUP / 32; ++it) {
      int j = it * 32 + l;
      float u = u0 + du * (float)j;
      int lo = 0, hi = C;                       // searchsorted(..., 'right')
      while (lo < hi) { int mid = (lo + hi) >> 1; if (cdfv[mid] <= u) lo = mid + 1; else hi = mid; }
      int below = (lo - 1 > 0) ? lo - 1 : 0;
      int above = (lo < C - 1) ? lo : C - 1;
      float c0 = cdfv[below], c1 = cdfv[above];
      float g0 = zmid[below], g1 = zmid[above];
      float den = c1 - c0; if (den < 1e-5f) den = 1.f;
      float tt = (u - c0) / den;
      znew[j] = g0 + tt * (g1 - g0);
    }
  }
  __syncthreads();

  // ---- fine samples + density ----
  for (int it = 0; it < NUP / 32; ++it) {
    int i = it * 32 + l;
    float z = znew[i];
    xs[i] = fminf(fmaxf(ox + dx * z, -1.f), 1.f);
    ys[i] = fminf(fmaxf(oy + dy * z, -1.f), 1.f);
    zs[i] = fminf(fmaxf(oz + dz * z, -1.f), 1.f);
  }
  __syncthreads();
  density_mlp_128(xs, ys, zs, sW1, sb1, sW2, sb2v, hbuf, sigma_src, geo, NRS);

  // ---- stable merge of two sorted lists (coarse wins ties, like argsort) ----
  for (int it = 0; it < NRS / 32; ++it) {
    int i = it * 32 + l;
    float v = zc[i];                            // coarse elem -> + count(znew < v)
    int lo = 0, hi = NUP;
    while (lo < hi) { int mid = (lo + hi) >> 1; if (znew[mid] < v) lo = mid + 1; else hi = mid; }
    int pos = i + lo;
    zall[pos] = v; srci[pos] = i;
    float w = znew[i];                          // fine elem -> + count(zc <= w)
    lo = 0; hi = NRS;
    while (lo < hi) { int mid = (lo + hi) >> 1; if (zc[mid] <= w) lo = mid + 1; else hi = mid; }
    pos = i + lo;
    zall[pos] = w; srci[pos] = NRS + i;
  }
  __syncthreads();

  // ---- final weights over 256 merged samples: chunked prefix product ----
  float wsum;
  {
    float fa[8], lp[8];
    float P = 1.f;
#pragma unroll
    for (int j = 0; j < 8; ++j) {
      int i = l * 8 + j;
      float delta = (i < NTOT - 1) ? (zall[i + 1] - zall[i]) : sd;
      float alpha = 1.f - expf(-delta * sigma_src[srci[i]]);
      fa[j] = alpha;
      lp[j] = P;
      P *= (1.f - alpha + 1e-15f);
    }
    float incl = wave_incl_scan_mul(P, l);
    float excl = __shfl_up(incl, 1, 32);
    if (l == 0) excl = 1.f;
    float lsum = 0.f;
#pragma unroll
    for (int j = 0; j < 8; ++j) {
      float w = fa[j] * excl * lp[j];
      wts[l * 8 + j] = w;
      lsum += w;
    }
#pragma unroll
    for (int off = 16; off > 0; off >>= 1) lsum += __shfl_xor(lsum, off, 32);
    wsum = lsum;                        // all lanes hold the full sum
  }
  __syncthreads();

  // ---- color MLP over 256 merged samples (16 M-tiles) + compositing ----
  float acc = 0.f;  // per-lane rgb partial (valid on lanes with m<3)
  for (int t = 0; t < 16; ++t) {
    const int p = t * 16 + m;
    const int s = srci[p];
    v8f h[4];
#pragma unroll
    for (int nt = 0; nt < 4; ++nt) {
      const float bias = sbc1[nt * 16 + m];
      v8f c;
#pragma unroll
      for (int r = 0; r < 8; ++r) c[r] = bias;
      h[nt] = c;
    }
#pragma unroll
    for (int ks = 0; ks < 5; ++ks) {    // K = 18 padded to 20
      const int k0 = ks * 4 + hf * 2;
      v2f a;
      {
        float a0 = (k0 == 0) ? dx : (k0 == 1) ? dy : (k0 == 2) ? dz
                   : (k0 < 18) ? geo[s * 16 + (k0 - 3)] : 0.f;
        int k1 = k0 + 1;
        float a1 = (k1 == 1) ? dy : (k1 == 2) ? dz
                   : (k1 < 18) ? geo[s * 16 + (k1 - 3)] : 0.f;
        a.x = a0; a.y = a1;
      }
#pragma unroll
      for (int nt = 0; nt < 4; ++nt) {
        v2f b;
        b.x = sWc1[(k0 + 0) * 64 + nt * 16 + m];
        b.y = sWc1[(k0 + 1) * 64 + nt * 16 + m];
        h[nt] = wmma4(a, b, h[nt]);
      }
    }
#pragma unroll
    for (int nt = 0; nt < 4; ++nt)
#pragma unroll
      for (int r = 0; r < 8; ++r)
        hbuf[(r + hf * 8) * 65 + nt * 16 + m] = fmaxf(h[nt][r], 0.f);
    __syncthreads();

    v8f o;
    {
      const float bias2 = sbc2v[m];
#pragma unroll
      for (int r = 0; r < 8; ++r) o[r] = bias2;
    }
#pragma unroll
    for (int ks = 0; ks < 16; ++ks) {
      v2f a2, b2;
      a2.x = hbuf[m * 65 + ks * 4 + hf * 2 + 0];
      a2.y = hbuf[m * 65 + ks * 4 + hf * 2 + 1];
      // Wc2 has only 3 real output columns; lanes m>=3 contribute zeros.
      b2.x = (m < 3) ? sWc2[(ks * 4 + hf * 2 + 0) * 4 + m] : 0.f;
      b2.y = (m < 3) ? sWc2[(ks * 4 + hf * 2 + 1) * 4 + m] : 0.f;
      o = wmma4(a2, b2, o);
    }
    if (m < 3) {                       // channel m, points M = r + hf*8
#pragma unroll
      for (int r = 0; r < 8; ++r) {
        int pp = t * 16 + r + hf * 8;
        acc += wts[pp] * sigmoid_f(o[r]);
      }
    }
    __syncthreads();
  }

  // combine the two half-wave partials; white background term
  acc += __shfl_xor(acc, 16, 32);
  if (l < 3) out[ray * 3 + l] = acc + (1.f - wsum);
}

}  // namespace

extern "C" void kernel_launch(void* const* d_in, const int* in_sizes, int n_in,
                              void* d_out, int out_size, void* d_ws, size_t ws_size,
                              hipStream_t stream) {
  const float* rays_o = (const float*)d_in[0];
  const float* rays_d = (const float*)d_in[1];
  const float* W1  = (const float*)d_in[2];
  const float* b1  = (const float*)d_in[3];
  const float* W2  = (const float*)d_in[4];
  const float* b2  = (const float*)d_in[5];
  const float* Wc1 = (const float*)d_in[6];
  const float* bc1 = (const float*)d_in[7];
  const float* Wc2 = (const float*)d_in[8];
  const float* bc2 = (const float*)d_in[9];
  const int n_rays = in_sizes[0] / 3;
  nerf_render_kernel<<<n_rays, 32, 0, stream>>>(rays_o, rays_d, W1, b1, W2, b2,
                                                Wc1, bc1, Wc2, bc2, (float*)d_out);
}